// RSA_layer_6193342841431
// MI455X (gfx1250) — compile-verified
//
#include <hip/hip_runtime.h>

#define UNITS  128
#define WINDOW 1024

typedef __attribute__((ext_vector_type(2))) float v2f;
typedef __attribute__((ext_vector_type(8))) float v8f;

// fs[j][u]: row j of the "new_state^T" matrix.
__device__ __forceinline__ float fs_elem(const float* __restrict__ state,
                                         const float* __restrict__ input,
                                         int j, int u) {
    return (j < WINDOW - 1) ? state[u * WINDOW + j + 1] : input[u];
}

// Kernel A:
//  blocks 0..63 : proj_hi = fs @ w_hi via V_WMMA_F32_16X16X4_F32
//                 (block b owns rows j0=16b; wave n owns cols 16n..16n+15)
//  block  64    : phj[u] = fs[1023] @ w_hj, dlast[j] = fs[j] . fs[1023]
__global__ void __launch_bounds__(256) rsa_projA(
    const float* __restrict__ input,
    const float* __restrict__ state,
    const float* __restrict__ w,      // (2U+1, U) row-major
    float* __restrict__ proj,         // (1024, 128)
    float* __restrict__ phj,          // (128)
    float* __restrict__ dlast)        // (1024)
{
    const int t = threadIdx.x;

    if (blockIdx.x < 64) {
        __shared__ float fs_lds[16][132];   // pad 128->132 (stride%64==4) to spread banks
        const int j0 = blockIdx.x * 16;

        // Stage 16x128 fs tile into LDS (coalesced along j within each u row of state).
        #pragma unroll
        for (int e = 0; e < 8; ++e) {
            int idx = t + 256 * e;          // 0..2047
            int u  = idx >> 4;
            int jj = idx & 15;
            fs_lds[jj][u] = fs_elem(state, input, j0 + jj, u);
        }
        __syncthreads();

        const int lane = t & 31;
        const int wid  = t >> 5;            // N-tile 0..7
        const int half = lane >> 4;         // 0: lanes 0-15, 1: lanes 16-31
        const int l15  = lane & 15;
        const int col  = wid * 16 + l15;    // output column (u)

        v8f acc = {0.f, 0.f, 0.f, 0.f, 0.f, 0.f, 0.f, 0.f};

        #pragma unroll 4
        for (int k = 0; k < 32; ++k) {      // K = 128 in steps of 4
            const int uk = 4 * k + 2 * half;
            v2f a, b;
            // A 16x4 f32: lanes 0-15 rows M, VGPR0 = {K=0 | K=2}, VGPR1 = {K=1 | K=3}
            a.x = fs_lds[l15][uk];
            a.y = fs_lds[l15][uk + 1];
            // B 4x16 f32: lanes hold N, VGPR0 = {K=0 | K=2}, VGPR1 = {K=1 | K=3}
            b.x = w[uk * UNITS + col];
            b.y = w[(uk + 1) * UNITS + col];
            acc = __builtin_amdgcn_wmma_f32_16x16x4_f32(
                      false, a, false, b, (short)0, acc, false, false);
        }

        // C/D 16x16 f32 layout: VGPR v -> M = v + 8*half, N = lane&15
        #pragma unroll
        for (int v = 0; v < 8; ++v) {
            int row = j0 + half * 8 + v;
            proj[row * UNITS + col] = acc[v];
        }
    } else {
        // Aux matvecs (tiny).
        if (t < UNITS) {
            float a = 0.f;
            for (int k = 0; k < UNITS; ++k)
                a += input[k] * w[(UNITS + k) * UNITS + t];
            phj[t] = a;
        }
        for (int j = t; j < WINDOW; j += 256) {
            float a = 0.f;
            for (int u = 0; u < UNITS; ++u)
                a += fs_elem(state, input, j, u) * input[u];
            dlast[j] = a;
        }
    }
}

// Kernel B: one block per u. sim[j] = proj[j,u] + phj[u] + dlast[j]*wdot[u] + b[u],
// softmax over j, out[u] = sum_j fs[j,u] * a[j].
__global__ void __launch_bounds__(256) rsa_softmax(
    const float* __restrict__ input,
    const float* __restrict__ state,
    const float* __restrict__ w,
    const float* __restrict__ bias,
    const float* __restrict__ proj,
    const float* __restrict__ phj,
    const float* __restrict__ dlast,
    float* __restrict__ out)
{
    const int u    = blockIdx.x;
    const int t    = threadIdx.x;
    const int lane = t & 31;
    const int wid  = t >> 5;

    __shared__ float red[8];
    __shared__ float redc[8];

    const float phj_u  = phj[u];
    const float wdot_u = w[2 * UNITS * UNITS + u];
    const float b_u    = bias[u];

    float sim[4], fsv[4];
    float lmax = -3.402823466e38f;
    #pragma unroll
    for (int r = 0; r < 4; ++r) {
        int j = t + 256 * r;
        fsv[r] = fs_elem(state, input, j, u);
        sim[r] = proj[j * UNITS + u] + phj_u + dlast[j] * wdot_u + b_u;
        lmax = fmaxf(lmax, sim[r]);
    }

    // block max (wave32 shuffle + LDS)
    #pragma unroll
    for (int o = 16; o > 0; o >>= 1) lmax = fmaxf(lmax, __shfl_xor(lmax, o, 32));
    if (lane == 0) red[wid] = lmax;
    __syncthreads();
    float gmax = red[0];
    #pragma unroll
    for (int i = 1; i < 8; ++i) gmax = fmaxf(gmax, red[i]);
    __syncthreads();

    float lsum = 0.f, lc = 0.f;
    #pragma unroll
    for (int r = 0; r < 4; ++r) {
        float e = __expf(sim[r] - gmax);
        lsum += e;
        lc   += e * fsv[r];
    }
    #pragma unroll
    for (int o = 16; o > 0; o >>= 1) {
        lsum += __shfl_xor(lsum, o, 32);
        lc   += __shfl_xor(lc,   o, 32);
    }
    if (lane == 0) { red[wid] = lsum; redc[wid] = lc; }
    __syncthreads();
    if (t == 0) {
        float Z = 0.f, C = 0.f;
        #pragma unroll
        for (int i = 0; i < 8; ++i) { Z += red[i]; C += redc[i]; }
        out[u] = C / Z;
    }
}

extern "C" void kernel_launch(void* const* d_in, const int* in_sizes, int n_in,
                              void* d_out, int out_size, void* d_ws, size_t ws_size,
                              hipStream_t stream) {
    const float* input = (const float*)d_in[0];   // (1,128)
    const float* state = (const float*)d_in[1];   // (128,1024)
    const float* w     = (const float*)d_in[2];   // (257,128)
    const float* bias  = (const float*)d_in[3];   // (128)

    float* proj  = (float*)d_ws;                  // 1024*128 f32
    float* phj   = proj + WINDOW * UNITS;         // 128 f32
    float* dlast = phj + UNITS;                   // 1024 f32

    rsa_projA<<<65, 256, 0, stream>>>(input, state, w, proj, phj, dlast);
    rsa_softmax<<<UNITS, 256, 0, stream>>>(input, state, w, bias,
                                           proj, phj, dlast, (float*)d_out);
}